// QuantumLatentGenerator_64183991271528
// MI455X (gfx1250) — compile-verified
//
#include <hip/hip_runtime.h>
#include <math.h>

// Quantum latent generator for MI455X (gfx1250, wave32).
// 1 workgroup == 1 batch element. Full 16384-amplitude complex state lives in
// LDS (~130 KB of the 320 KB WGP LDS) as interleaved float2 (re,im) so every
// amplitude moves as one ds_load_b64/ds_store_b64. All 70 gates applied
// in-LDS with barriers; CZ sign layers fused into the wire-13 gate of each
// depth layer. Final probs @ Z_SIGNS^T projection uses V_WMMA_F32_16X16X4_F32
// via the factorization idx = 128*hi + lo (GEMM 128x128 @ 128x16 per batch);
// +-1 sign fragments generated branchlessly by sign-bit injection, A
// fragments loaded as single b128 (two interleaved amplitudes).

#define NQ     14
#define DIM    (1 << NQ)          // 16384
#define BATCH  256
#define DEPTH  4
#define NGATES (NQ + DEPTH * NQ)  // 70
#define BLOCK  512                // 16 waves of 32

typedef __attribute__((ext_vector_type(2))) float v2f;
typedef __attribute__((ext_vector_type(8))) float v8f;

__device__ __forceinline__ float czsign(int idx) {
    // parity of adjacent-bit products == popcount(idx & (idx>>1)) & 1
    return 1.0f - 2.0f * (float)(__popc(idx & (idx >> 1)) & 1);
}

__global__ void __launch_bounds__(BLOCK)
qlatent_kernel(const float* __restrict__ cond,
               const float* __restrict__ W_enc,
               const float* __restrict__ b_enc,
               const float* __restrict__ weights,
               float* __restrict__ out)
{
    extern __shared__ float smem[];
    float2* st  = (float2*)smem;       // [DIM] interleaved (re, im)
    float*  gU  = smem + 2 * DIM;      // [NGATES*8] 2x2 complex gates
    float*  lat = gU + NGATES * 8;     // [16] latent accumulators

    const int b   = blockIdx.x;
    const int tid = threadIdx.x;

    // ---- init state |0...0>, clear latent accumulators ----
    for (int i = tid; i < DIM; i += BLOCK) st[i] = make_float2(0.0f, 0.0f);
    if (tid == 0)  st[0] = make_float2(1.0f, 0.0f);
    if (tid < 16)  lat[tid] = 0.0f;

    // ---- precompute all 70 gate matrices into LDS ----
    if (tid < NQ) {
        // encoding gate: U = RY(v) @ RX(v), v = silu(cond @ W_enc^T + b_enc)[tid]
        float acc = b_enc[tid];
        for (int j = 0; j < NQ; ++j) acc += cond[b * NQ + j] * W_enc[tid * NQ + j];
        float e = acc / (1.0f + expf(-acc));      // silu
        float h = 0.5f * e;
        float c = cosf(h), s = sinf(h);
        float* u = &gU[tid * 8];
        u[0] =  c * c; u[1] =  s * s;   // U00 = c^2 + i s^2
        u[2] = -s * c; u[3] = -s * c;   // U01 = -sc - i sc
        u[4] =  s * c; u[5] = -s * c;   // U10 =  sc - i sc
        u[6] =  c * c; u[7] = -s * s;   // U11 = c^2 - i s^2
    } else if (tid < NGATES) {
        // layer gate: U = RY(p2) @ RX(p1) @ RZ(p0)
        int d = (tid - NQ) / NQ, w = (tid - NQ) % NQ;
        const float* p = &weights[(d * NQ + w) * 3];
        float c1 = cosf(0.5f * p[1]), s1 = sinf(0.5f * p[1]);
        float c2 = cosf(0.5f * p[2]), s2 = sinf(0.5f * p[2]);
        // M = RY(p2) @ RX(p1)
        float M00r =  c2 * c1, M00i =  s2 * s1;
        float M01r = -s2 * c1, M01i = -c2 * s1;
        float M10r =  s2 * c1, M10i = -c2 * s1;
        float M11r =  c2 * c1, M11i = -s2 * s1;
        float cz = cosf(0.5f * p[0]), sz = sinf(0.5f * p[0]);
        // col 0 scaled by e^{-i p0/2} = (cz, -sz); col 1 by (cz, +sz)
        float* u = &gU[tid * 8];
        u[0] = M00r * cz + M00i * sz;  u[1] = M00i * cz - M00r * sz;
        u[2] = M01r * cz - M01i * sz;  u[3] = M01i * cz + M01r * sz;
        u[4] = M10r * cz + M10i * sz;  u[5] = M10i * cz - M10r * sz;
        u[6] = M11r * cz - M11i * sz;  u[7] = M11i * cz + M11r * sz;
    }
    __syncthreads();

    // ---- apply all gates in LDS; fuse CZ signs into wire-13 gate of each layer ----
    for (int g = 0; g < NGATES; ++g) {
        int wire = (g < NQ) ? g : (g - NQ) % NQ;
        int k = NQ - 1 - wire;                 // LSB bit position of this qubit
        float u00r = gU[g * 8 + 0], u00i = gU[g * 8 + 1];
        float u01r = gU[g * 8 + 2], u01i = gU[g * 8 + 3];
        float u10r = gU[g * 8 + 4], u10i = gU[g * 8 + 5];
        float u11r = gU[g * 8 + 6], u11i = gU[g * 8 + 7];
        bool applySigns = (g >= NQ) && (wire == NQ - 1);
        int mask = (1 << k) - 1;
        for (int t = tid; t < (DIM >> 1); t += BLOCK) {
            int i0 = ((t & ~mask) << 1) | (t & mask);
            int i1 = i0 | (1 << k);
            float2 A = st[i0];                 // one ds_load_b64 each
            float2 B = st[i1];
            float nar = u00r * A.x - u00i * A.y + u01r * B.x - u01i * B.y;
            float nai = u00r * A.y + u00i * A.x + u01r * B.y + u01i * B.x;
            float nbr = u10r * A.x - u10i * A.y + u11r * B.x - u11i * B.y;
            float nbi = u10r * A.y + u10i * A.x + u11r * B.y + u11i * B.x;
            if (applySigns) {
                float s0 = czsign(i0), s1 = czsign(i1);
                nar *= s0; nai *= s0; nbr *= s1; nbi *= s1;
            }
            st[i0] = make_float2(nar, nai);    // one ds_store_b64 each
            st[i1] = make_float2(nbr, nbi);
        }
        __syncthreads();
    }

    // ---- projection: latent = probs @ Z_SIGNS^T via f32 WMMA ----
    // idx = 128*hi + lo.  T[hi][n] = sum_lo P[hi][lo] * SL[lo][n]   (WMMA GEMM)
    // latent[n] = sum_hi w_n(hi) * T[hi][n]                         (reduction)
    // 16 waves: wave (2m+kh) computes M-tile m over K-half kh.
    {
        const int lane    = threadIdx.x & 31;
        const int waveId  = threadIdx.x >> 5;        // wave32
        const int m       = waveId >> 1;             // M-tile (rows hi = 16m..16m+15)
        const int kh      = waveId & 1;              // K half: kk in [16*kh, 16*kh+16)
        const int halfSel = (lane < 16) ? 0 : 1;
        const int row     = m * 16 + (lane & 15);    // A-fragment row
        const int n       = lane & 15;               // B/D column

        // Branchless +-1 fragment generators (sign-bit injection, per-lane consts):
        // B side (columns 7..13 take sign from bit (13-n) of lo; 0..6 -> 1; 14..15 -> 0)
        const int      bmsel = (n >= 7 && n <= 13) ? (1 << (13 - n)) : 0;
        const int      bsh   = (18 + n > 31) ? 31 : (18 + n);   // bit (13-n) -> bit 31
        const unsigned bbase = (n <= 13) ? 0x3F800000u : 0u;    // 1.0f or 0.0f
        // hi-side weight (columns 0..6 take sign from bit (6-n) of hi; 7..13 -> 1)
        const int      wmsel = (n <= 6) ? (1 << (6 - n)) : 0;
        const int      wsh   = (25 + n > 31) ? 31 : (25 + n);   // bit (6-n) -> bit 31
        const unsigned wbase = (n <= 13) ? 0x3F800000u : 0u;

        const float4* st4 = (const float4*)smem;     // two interleaved amplitudes

        v8f accv = {};
        #pragma unroll
        for (int kk = 0; kk < 16; ++kk) {
            int k0 = (kh * 16 + kk) * 4;
            int ka = k0 + halfSel * 2;       // A: K = {0,1} or {2,3} per lane half
            int ia = row * 128 + ka;         // even -> 16B aligned pair
            float4 amp = st4[ia >> 1];       // one ds_load_b128: {re0,im0,re1,im1}
            v2f a;  a.x = amp.x * amp.x + amp.y * amp.y;   // prob[row][ka]
                    a.y = amp.z * amp.z + amp.w * amp.w;   // prob[row][ka+1]
            int lo0 = ka, lo1 = ka + 1;
            v2f bf;
            bf.x = __uint_as_float(bbase ^ (((unsigned)(lo0 & bmsel)) << bsh));
            bf.y = __uint_as_float(bbase ^ (((unsigned)(lo1 & bmsel)) << bsh));
            accv = __builtin_amdgcn_wmma_f32_16x16x4_f32(
                       false, a, false, bf, (short)0, accv, false, false);
        }
        // D layout: VGPR r holds M = r (lanes 0-15) or r+8 (lanes 16-31)
        float partial = 0.0f;
        #pragma unroll
        for (int r = 0; r < 8; ++r) {
            int hi = m * 16 + r + halfSel * 8;
            float w = __uint_as_float(wbase ^ (((unsigned)(hi & wmsel)) << wsh));
            partial += w * accv[r];
        }
        atomicAdd(&lat[n], partial);         // ds_add_f32
    }
    __syncthreads();

    // ---- layernorm over 14 latent values, write out ----
    if (tid == 0) {
        float mu = 0.0f;
        for (int l = 0; l < NQ; ++l) mu += lat[l];
        mu *= (1.0f / NQ);
        float var = 0.0f;
        for (int l = 0; l < NQ; ++l) { float d = lat[l] - mu; var += d * d; }
        var *= (1.0f / NQ);
        float inv = rsqrtf(var + 1e-5f);
        for (int l = 0; l < NQ; ++l) out[b * NQ + l] = (lat[l] - mu) * inv;
    }
}

extern "C" void kernel_launch(void* const* d_in, const int* in_sizes, int n_in,
                              void* d_out, int out_size, void* d_ws, size_t ws_size,
                              hipStream_t stream) {
    (void)in_sizes; (void)n_in; (void)d_ws; (void)ws_size; (void)out_size;
    const float* cond  = (const float*)d_in[0];   // (256,14)
    const float* W_enc = (const float*)d_in[1];   // (14,14)
    const float* b_enc = (const float*)d_in[2];   // (14,)
    const float* wts   = (const float*)d_in[3];   // (4,14,3)
    float* out = (float*)d_out;                   // (256,14) f32

    size_t shmem = (size_t)(2 * DIM + NGATES * 8 + 16) * sizeof(float); // ~130.3 KB
    qlatent_kernel<<<dim3(BATCH), dim3(BLOCK), shmem, stream>>>(cond, W_enc, b_enc, wts, out);
}